// convEP_86552180949477
// MI455X (gfx1250) — compile-verified
//
#include <hip/hip_runtime.h>
#include <hip/hip_bf16.h>

typedef float v2f __attribute__((ext_vector_type(2)));
typedef float v8f __attribute__((ext_vector_type(8)));

__device__ __forceinline__ v8f wmma_f32(v2f a, v2f b, v8f c) {
  // V_WMMA_F32_16X16X4_F32 : D = A(16x4) * B(4x16) + C(16x16), all fp32
  return __builtin_amdgcn_wmma_f32_16x16x4_f32(
      /*neg_a=*/false, a, /*neg_b=*/false, b,
      /*c_mod=*/(short)0, c, /*reuse_a=*/false, /*reuse_b=*/false);
}

__device__ __forceinline__ float rho_(float v) { return fminf(fmaxf(v, 0.f), 1.f); }

// ---------------------------------------------------------------------------
// conv0 (5x5, pad 2) over s2[64,64,16,16] with w[128,64,5,5] + bias, then
// 2x2 maxpool with argmax.  Implicit GEMM: M = row of 16 pixels, N = 16 couts,
// K = Cin*25 in chunks of 4 via V_WMMA_F32_16X16X4_F32.
// Input tile staged zero-padded in LDS once -> branch-free inner loop.
// grid = (8 cout-groups, 64 batch), block = 256 (wave w -> rows 2w, 2w+1)
// dyn LDS = 5*16*64 + 20*20*66 floats = 126,080 B
// ---------------------------------------------------------------------------
#define C0_CI   64
#define C0_CPAD 66
__global__ void conv0_pool_wmma(const float* __restrict__ s2,
                                const float* __restrict__ w,
                                const float* __restrict__ bias,
                                float* __restrict__ p1out,
                                int* __restrict__ idxout) {
  extern __shared__ float smem[];
  float* wtile = smem;                    // [kx][co][ci]  : 5*16*64
  float* inpad = smem + 5 * 16 * C0_CI;   // [y][x][cipad] : 20*20*66

  const int b    = blockIdx.y;
  const int cog  = blockIdx.x;
  const int lane = threadIdx.x & 31;
  const int wave = threadIdx.x >> 5;
  const int nidx = lane & 15;
  const int kh   = lane >> 4;
  const int y0   = wave * 2, y1 = y0 + 1;

  // stage zero-padded input tile (halo handled once, not in the K loop)
  for (int g = threadIdx.x; g < 20 * 20 * C0_CI; g += 256) {
    const int x = g % 20, y = (g / 20) % 20, ci = g / 400;
    const int iy = y - 2, ix = x - 2;
    float v = 0.f;
    if ((unsigned)iy < 16u && (unsigned)ix < 16u)
      v = s2[((b * C0_CI + ci) * 16 + iy) * 16 + ix];
    inpad[(y * 20 + x) * C0_CPAD + ci] = v;
  }

  v8f acc0 = {0.f, 0.f, 0.f, 0.f, 0.f, 0.f, 0.f, 0.f};
  v8f acc1 = {0.f, 0.f, 0.f, 0.f, 0.f, 0.f, 0.f, 0.f};

  const int xb = lane & 15;
  for (int ky = 0; ky < 5; ++ky) {
    __syncthreads();
    for (int t = threadIdx.x; t < 5 * 16 * C0_CI; t += 256) {
      const int ci = t & 63, co = (t >> 6) & 15, kx = t >> 10;
      wtile[t] = w[(((cog * 16 + co) * C0_CI + ci) * 5 + ky) * 5 + kx];
    }
    __syncthreads();
    for (int kx = 0; kx < 5; ++kx) {
      const float* arow0 = &inpad[((y0 + ky) * 20 + xb + kx) * C0_CPAD + kh * 2];
      const float* arow1 = &inpad[((y1 + ky) * 20 + xb + kx) * C0_CPAD + kh * 2];
      const float* brow  = &wtile[(kx * 16 + nidx) * C0_CI + kh * 2];
#pragma unroll
      for (int kc = 0; kc < 16; ++kc) {      // Cin=64 in chunks of 4
        v2f af0 = *(const v2f*)(arow0 + kc * 4);
        v2f af1 = *(const v2f*)(arow1 + kc * 4);
        v2f bf  = *(const v2f*)(brow  + kc * 4);
        acc0 = wmma_f32(af0, bf, acc0);
        acc1 = wmma_f32(af1, bf, acc1);
      }
    }
  }

  // epilogue: +bias, 2x2 maxpool with first-occurrence argmax (order 2*dr+dc)
  const float bv = bias[cog * 16 + nidx];
#pragma unroll
  for (int p = 0; p < 4; ++p) {
    float c0 = acc0[2 * p]     + bv;   // (dr=0,dc=0)
    float c1 = acc0[2 * p + 1] + bv;   // (0,1)
    float c2 = acc1[2 * p]     + bv;   // (1,0)
    float c3 = acc1[2 * p + 1] + bv;   // (1,1)
    float m = c0; int best = 0;
    if (c1 > m) { m = c1; best = 1; }
    if (c2 > m) { m = c2; best = 2; }
    if (c3 > m) { m = c3; best = 3; }
    const int px = p + 4 * kh;
    const int o  = ((b * 128 + cog * 16 + nidx) * 8 + wave) * 8 + px;
    p1out[o]  = m;
    idxout[o] = best;
  }
}

// ---------------------------------------------------------------------------
// transpose-conv fused with unpool:  s2' = rho(p2 + conv(unpool(s1_old,idx), tw))
// tw[co][ci][ky][kx] = conv0_w[ci][co][4-ky][4-kx].  The unpool is evaluated on
// the fly while staging the padded input tile in LDS (no u buffer in global).
// grid = (4 cout-groups, 64 batch), block = 256
// dyn LDS = 5*16*128 + 20*20*130 floats = 248,960 B  (<= 320KB/WGP)
// ---------------------------------------------------------------------------
#define TC_CI   128
#define TC_CPAD 130
__global__ void tconv_s2_wmma(const float* __restrict__ s1o,
                              const int* __restrict__ idx,
                              const float* __restrict__ w0,
                              const float* __restrict__ p2,
                              float* __restrict__ s2n) {
  extern __shared__ float smem[];
  float* wtile = smem;                    // [kx][co][ci]  : 5*16*128
  float* inpad = smem + 5 * 16 * TC_CI;   // [y][x][cipad] : 20*20*130

  const int b    = blockIdx.y;
  const int cog  = blockIdx.x;
  const int lane = threadIdx.x & 31;
  const int wave = threadIdx.x >> 5;
  const int nidx = lane & 15;
  const int kh   = lane >> 4;
  const int y0   = wave * 2, y1 = y0 + 1;

  // stage padded unpool(s1_old, idx) tile: u[ci, iy, ix] reconstructed on the fly
  for (int g = threadIdx.x; g < 20 * 20 * TC_CI; g += 256) {
    const int x = g % 20, y = (g / 20) % 20, ci = g / 400;
    const int iy = y - 2, ix = x - 2;
    float v = 0.f;
    if ((unsigned)iy < 16u && (unsigned)ix < 16u) {
      const int py = iy >> 1, dr = iy & 1, px = ix >> 1, dc = ix & 1;
      const int pi = ((b * TC_CI + ci) * 8 + py) * 8 + px;
      v = (idx[pi] == dr * 2 + dc) ? s1o[pi] : 0.f;
    }
    inpad[(y * 20 + x) * TC_CPAD + ci] = v;
  }

  v8f acc0 = {0.f, 0.f, 0.f, 0.f, 0.f, 0.f, 0.f, 0.f};
  v8f acc1 = {0.f, 0.f, 0.f, 0.f, 0.f, 0.f, 0.f, 0.f};

  const int xb = lane & 15;
  for (int ky = 0; ky < 5; ++ky) {
    __syncthreads();
    for (int t = threadIdx.x; t < 5 * 16 * TC_CI; t += 256) {
      const int ci = t & 127, co = (t >> 7) & 15, kx = t >> 11;
      wtile[t] = w0[((ci * 64 + (cog * 16 + co)) * 5 + (4 - ky)) * 5 + (4 - kx)];
    }
    __syncthreads();
    for (int kx = 0; kx < 5; ++kx) {
      const float* arow0 = &inpad[((y0 + ky) * 20 + xb + kx) * TC_CPAD + kh * 2];
      const float* arow1 = &inpad[((y1 + ky) * 20 + xb + kx) * TC_CPAD + kh * 2];
      const float* brow  = &wtile[(kx * 16 + nidx) * TC_CI + kh * 2];
#pragma unroll
      for (int kc = 0; kc < 32; ++kc) {      // Cin=128 in chunks of 4
        v2f af0 = *(const v2f*)(arow0 + kc * 4);
        v2f af1 = *(const v2f*)(arow1 + kc * 4);
        v2f bf  = *(const v2f*)(brow  + kc * 4);
        acc0 = wmma_f32(af0, bf, acc0);
        acc1 = wmma_f32(af1, bf, acc1);
      }
    }
  }

#pragma unroll
  for (int v = 0; v < 8; ++v) {
    const int x  = v + 8 * kh;
    const int o0 = ((b * 64 + cog * 16 + nidx) * 16 + y0) * 16 + x;
    const int o1 = o0 + 16;
    s2n[o0] = rho_(p2[o0] + acc0[v]);
    s2n[o1] = rho_(p2[o1] + acc1[v]);
  }
}

// p2 = maxpool2x2(conv2d(data, conv1_w) + b1) — loop invariant, computed once.
__global__ void p2_precompute(const float* __restrict__ data,
                              const float* __restrict__ w1,
                              const float* __restrict__ b1,
                              float* __restrict__ p2) {
  const int g  = blockIdx.x * blockDim.x + threadIdx.x;   // 1,048,576
  const int px = g & 15, py = (g >> 4) & 15, co = (g >> 8) & 63, b = g >> 14;
  float m = -3.4e38f;
#pragma unroll
  for (int dr = 0; dr < 2; ++dr)
#pragma unroll
    for (int dc = 0; dc < 2; ++dc) {
      const int y = 2 * py + dr, x = 2 * px + dc;
      float acc = b1[co];
      for (int ci = 0; ci < 3; ++ci)
        for (int ky = 0; ky < 5; ++ky)
          for (int kx = 0; kx < 5; ++kx) {
            const int yy = y + ky - 2, xx = x + kx - 2;
            if ((unsigned)yy < 32u && (unsigned)xx < 32u)
              acc += data[((b * 3 + ci) * 32 + yy) * 32 + xx] *
                     w1[((co * 3 + ci) * 5 + ky) * 5 + kx];
          }
      m = fmaxf(m, acc);
    }
  p2[g] = m;
}

// s0' = rho(s1_flat @ fc_w^T + fc_b) : one wave per (b, o), K=8192 reduction
__global__ void fc_fwd(const float* __restrict__ s1o,
                       const float* __restrict__ fcw,
                       const float* __restrict__ fcb,
                       float* __restrict__ s0n) {
  const int lane = threadIdx.x & 31, wave = threadIdx.x >> 5;
  const int g = blockIdx.x * 8 + wave;           // 80 blocks * 8 waves = 640
  const int b = g / 10, o = g - b * 10;
  const float* a  = s1o + b * 8192;
  const float* wr = fcw + o * 8192;
  float s = 0.f;
  for (int i = lane; i < 8192; i += 32) s += a[i] * wr[i];
  for (int off = 16; off; off >>= 1) s += __shfl_xor(s, off, 32);
  if (lane == 0) s0n[b * 10 + o] = rho_(s + fcb[o]);
}

// s1' = rho(p1 + (s0 @ fc_w))  : one thread per element of [64,8192]
__global__ void fc_td_s1(const float* __restrict__ s0o,
                         const float* __restrict__ fcw,
                         const float* __restrict__ p1,
                         float* __restrict__ s1n) {
  const int g = blockIdx.x * blockDim.x + threadIdx.x;    // 524,288
  const int b = g >> 13, i = g & 8191;
  float td = 0.f;
#pragma unroll
  for (int j = 0; j < 10; ++j) td += s0o[b * 10 + j] * fcw[j * 8192 + i];
  s1n[g] = rho_(p1[g] + td);
}

extern "C" void kernel_launch(void* const* d_in, const int* in_sizes, int n_in,
                              void* d_out, int out_size, void* d_ws, size_t ws_size,
                              hipStream_t stream) {
  (void)in_sizes; (void)n_in; (void)out_size; (void)ws_size;
  const float* data    = (const float*)d_in[0];
  const float* s0_in   = (const float*)d_in[1];
  const float* s1_in   = (const float*)d_in[2];
  const float* s2_in   = (const float*)d_in[3];
  const float* conv0_w = (const float*)d_in[4];
  const float* conv0_b = (const float*)d_in[5];
  const float* conv1_w = (const float*)d_in[6];
  const float* conv1_b = (const float*)d_in[7];
  const float* fc_w    = (const float*)d_in[8];
  const float* fc_b    = (const float*)d_in[9];
  float* out = (float*)d_out;

  const int N_S0 = 64 * 10;               // 640
  const int N_S1 = 64 * 128 * 8 * 8;      // 524288
  const int N_S2 = 64 * 64 * 16 * 16;     // 1048576

  // workspace layout (floats/ints, all 4-byte)
  float* ws  = (float*)d_ws;
  float* p2  = ws;                        // N_S2
  float* p1  = p2 + N_S2;                 // N_S1
  int*   idx = (int*)(p1 + N_S1);         // N_S1 ints
  float* s0a = (float*)(idx + N_S1); float* s0b = s0a + N_S0;
  float* s1a = s0b + N_S0;           float* s1b = s1a + N_S1;
  float* s2a = s1b + N_S1;           float* s2b = s2a + N_S2;

  // init states
  hipMemcpyAsync(s0a, s0_in, N_S0 * sizeof(float), hipMemcpyDeviceToDevice, stream);
  hipMemcpyAsync(s1a, s1_in, N_S1 * sizeof(float), hipMemcpyDeviceToDevice, stream);
  hipMemcpyAsync(s2a, s2_in, N_S2 * sizeof(float), hipMemcpyDeviceToDevice, stream);

  // loop-invariant bottom-up path
  p2_precompute<<<N_S2 / 256, 256, 0, stream>>>(data, conv1_w, conv1_b, p2);

  const size_t lds_conv0 = (5 * 16 * C0_CI + 20 * 20 * C0_CPAD) * sizeof(float);
  const size_t lds_tconv = (5 * 16 * TC_CI + 20 * 20 * TC_CPAD) * sizeof(float);

  for (int t = 0; t < 10; ++t) {
    const bool cur = (t & 1) != 0;
    const float* s0o = cur ? s0b : s0a;  float* s0n = cur ? s0a : s0b;
    const float* s1o = cur ? s1b : s1a;  float* s1n = cur ? s1a : s1b;
    const float* s2o = cur ? s2b : s2a;  float* s2n = cur ? s2a : s2b;

    conv0_pool_wmma<<<dim3(8, 64), 256, lds_conv0, stream>>>(s2o, conv0_w, conv0_b, p1, idx);
    fc_fwd<<<80, 256, 0, stream>>>(s1o, fc_w, fc_b, s0n);
    fc_td_s1<<<N_S1 / 256, 256, 0, stream>>>(s0o, fc_w, p1, s1n);
    tconv_s2_wmma<<<dim3(4, 64), 256, lds_tconv, stream>>>(s1o, idx, conv0_w, p2, s2n);

    if (t == 6) {  // snapshot post-update states (st0, st1, st2)
      hipMemcpyAsync(out + N_S0 + N_S1 + N_S2,               s0n, N_S0 * sizeof(float), hipMemcpyDeviceToDevice, stream);
      hipMemcpyAsync(out + N_S0 + N_S1 + N_S2 + N_S0,        s1n, N_S1 * sizeof(float), hipMemcpyDeviceToDevice, stream);
      hipMemcpyAsync(out + N_S0 + N_S1 + N_S2 + N_S0 + N_S1, s2n, N_S2 * sizeof(float), hipMemcpyDeviceToDevice, stream);
    }
  }

  // final states: after t=9 the "new" buffers are the a-set
  hipMemcpyAsync(out,               s0a, N_S0 * sizeof(float), hipMemcpyDeviceToDevice, stream);
  hipMemcpyAsync(out + N_S0,        s1a, N_S1 * sizeof(float), hipMemcpyDeviceToDevice, stream);
  hipMemcpyAsync(out + N_S0 + N_S1, s2a, N_S2 * sizeof(float), hipMemcpyDeviceToDevice, stream);
}